// SetGather_74603581931600
// MI455X (gfx1250) — compile-verified
//
#include <hip/hip_runtime.h>
#include <hip/hip_bf16.h>
#include <math.h>

// ---------------------------------------------------------------------------
// Set2Set-style segment attention + LSTM for MI455X (gfx1250, wave32).
//   - attention: 1 workgroup / molecule, atoms cached in LDS (single HBM pass)
//   - LSTM GEMM: V_WMMA_F32_16X16X4_F32; 1 wave computes a 32x16 tile of all
//     4 gates (A shared across gates, B shared across the two 16-row tiles),
//     with running-pointer addressing so B loads use immediate offsets.
// ---------------------------------------------------------------------------

typedef float v2f __attribute__((ext_vector_type(2)));
typedef float v8f __attribute__((ext_vector_type(8)));

#define CACHE_ATOMS 112   // 112*128*4 = 56 KB LDS row cache
#define MAX_SC      512   // max atoms per molecule handled (mean 64, sd ~8)

__device__ __forceinline__ float sigmoidf_(float x) {
    return 1.0f / (1.0f + __expf(-x));
}

// --------------------------- segment starts --------------------------------
__global__ void seg_starts_kernel(const int* __restrict__ idx, int N, int Bp1,
                                  int* __restrict__ seg_start) {
    int b = blockIdx.x * blockDim.x + threadIdx.x;
    if (b >= Bp1) return;
    // lower_bound: first i with idx[i] >= b  (idx is sorted)
    int lo = 0, hi = N;
    while (lo < hi) {
        int mid = (lo + hi) >> 1;
        if (idx[mid] < b) lo = mid + 1; else hi = mid;
    }
    seg_start[b] = lo;
}

// --------------------------- zero init -------------------------------------
__global__ void zero_kernel(float* __restrict__ p, size_t n) {
    size_t i = (size_t)blockIdx.x * blockDim.x + threadIdx.x;
    if (i < n) p[i] = 0.0f;
}

// --------------------------- attention (D == 128) --------------------------
// scores -> segment softmax -> readout; writes E[b] = {carry[b], readout[b]}
__global__ __launch_bounds__(128) void attn_kernel(
    const float* __restrict__ af,        // [N,128]
    const int*   __restrict__ seg_start, // [B+1]
    const float* __restrict__ carry,     // [B,128]
    float*       __restrict__ E)         // [B,256]
{
    __shared__ float sCarry[128];
    __shared__ float sCache[CACHE_ATOMS * 128];
    __shared__ float sSc[MAX_SC];
    __shared__ float sRed[128];

    const int b    = blockIdx.x;
    const int tid  = threadIdx.x;
    const int wave = tid >> 5;
    const int lane = tid & 31;

    const int start = seg_start[b];
    int cnt = seg_start[b + 1] - start;
    if (cnt > MAX_SC) cnt = MAX_SC;   // astronomically unlikely overflow

    sCarry[tid] = carry[(size_t)b * 128 + tid];
    __syncthreads();

    const float c0 = sCarry[lane * 4 + 0];
    const float c1 = sCarry[lane * 4 + 1];
    const float c2 = sCarry[lane * 4 + 2];
    const float c3 = sCarry[lane * 4 + 3];

    // ---- phase 1: stream rows once; cache in LDS; per-wave dot products ----
    for (int a = wave; a < cnt; a += 4) {
        const float* rowp = af + (size_t)(start + a) * 128;
        if (a + 8 < cnt)  // stays ~4 KB ahead of the stream
            __builtin_prefetch(af + (size_t)(start + a + 8) * 128 + lane * 4, 0, 0);
        const float4 v = *(const float4*)(rowp + lane * 4);
        if (a < CACHE_ATOMS)
            *(float4*)(sCache + a * 128 + lane * 4) = v;
        float s = v.x * c0 + v.y * c1 + v.z * c2 + v.w * c3;
        s += __shfl_xor(s, 16, 32);
        s += __shfl_xor(s,  8, 32);
        s += __shfl_xor(s,  4, 32);
        s += __shfl_xor(s,  2, 32);
        s += __shfl_xor(s,  1, 32);
        if (lane == 0) sSc[a] = s;
    }
    __syncthreads();

    // ---- phase 2: segment softmax (max, then exp-sum) ----
    float m = -3.402823466e38f;
    for (int i = tid; i < cnt; i += 128) m = fmaxf(m, sSc[i]);
    sRed[tid] = m;
    __syncthreads();
    for (int off = 64; off > 0; off >>= 1) {
        if (tid < off) sRed[tid] = fmaxf(sRed[tid], sRed[tid + off]);
        __syncthreads();
    }
    const float mx = sRed[0];
    __syncthreads();

    float lsum = 0.0f;
    for (int i = tid; i < cnt; i += 128) {
        float e = __expf(sSc[i] - mx);
        sSc[i] = e;
        lsum += e;
    }
    sRed[tid] = lsum;
    __syncthreads();
    for (int off = 64; off > 0; off >>= 1) {
        if (tid < off) sRed[tid] += sRed[tid + off];
        __syncthreads();
    }
    const float den = sRed[0];
    __syncthreads();

    // ---- phase 3: readout column tid, mostly from LDS ----
    float acc = 0.0f;
    for (int a = 0; a < cnt; ++a) {
        const float w = sSc[a];
        const float x = (a < CACHE_ATOMS)
                            ? sCache[a * 128 + tid]
                            : af[(size_t)(start + a) * 128 + tid];
        acc += w * x;
    }
    const float ro = (den > 0.0f) ? (acc / den) : 0.0f;

    E[(size_t)b * 256 + tid]       = sCarry[tid];
    E[(size_t)b * 256 + 128 + tid] = ro;
}

// --------------------------- LSTM cell via WMMA ----------------------------
// One wave per (32-row, 16-col) tile of D. Per k-step: 2 A-frag loads,
// 8 B-frag loads (immediate offsets off one running pointer), 8 WMMAs.
__global__ __launch_bounds__(32) void lstm_kernel(
    const float* __restrict__ E,     // [B, 2D]
    const float* __restrict__ W,     // [2D, 4D] row-major
    const float* __restrict__ bias,  // [4D]
    float*       __restrict__ mem,   // [B, D]
    float*       __restrict__ carry, // [B, D]
    int B, int D)
{
    const int lane = threadIdx.x & 31;
    const int half = lane >> 4;      // 0: K+{0,1} / M=v ; 1: K+{2,3} / M=v+8
    const int l16  = lane & 15;
    const int m0   = blockIdx.x * 32;
    const int j0   = blockIdx.y * 16;     // column within D
    const int K    = 2 * D;               // 256
    const int N4   = 4 * D;               // 512 (W row stride)

    v8f accv[2][4];
#pragma unroll
    for (int t = 0; t < 2; ++t)
#pragma unroll
        for (int g = 0; g < 4; ++g)
#pragma unroll
            for (int v = 0; v < 8; ++v) accv[t][g][v] = 0.0f;

    // A-matrix: lane -> row M, regs hold K; running pointers, imm offsets
    const float* ap0 = E + (size_t)(m0 + l16) * K + 2 * half;
    const float* ap1 = ap0 + (size_t)16 * K;
    const float* wp  = W + (size_t)(2 * half) * N4 + j0 + l16;

#pragma unroll 4
    for (int k = 0; k < K; k += 4) {
        v2f a0; a0.x = ap0[0]; a0.y = ap0[1];
        v2f a1; a1.x = ap1[0]; a1.y = ap1[1];
#pragma unroll
        for (int g = 0; g < 4; ++g) {
            v2f bf; bf.x = wp[g * D]; bf.y = wp[g * D + N4];
            accv[0][g] = __builtin_amdgcn_wmma_f32_16x16x4_f32(
                false, a0, false, bf, (short)0, accv[0][g], false, false);
            accv[1][g] = __builtin_amdgcn_wmma_f32_16x16x4_f32(
                false, a1, false, bf, (short)0, accv[1][g], false, false);
        }
        ap0 += 4; ap1 += 4; wp += (size_t)4 * N4;
    }

    float bcol[4];
#pragma unroll
    for (int g = 0; g < 4; ++g) bcol[g] = bias[g * D + j0 + l16];

    const int col = j0 + l16;
#pragma unroll
    for (int t = 0; t < 2; ++t) {
#pragma unroll
        for (int v = 0; v < 8; ++v) {
            const int row = m0 + t * 16 + v + half * 8;  // C/D: lane>=16 -> M=v+8
            const size_t p = (size_t)row * D + col;
            const float u = sigmoidf_(accv[t][0][v] + bcol[0]);
            const float f = sigmoidf_(accv[t][1][v] + bcol[1]);
            const float c = tanhf(accv[t][2][v] + bcol[2]);
            const float o = sigmoidf_(accv[t][3][v] + bcol[3]);
            const float mo = mem[p];
            const float mn = f * mo + u * c;
            mem[p]   = mn;
            carry[p] = o * tanhf(mn);
        }
    }
}

// --------------------------- launch ----------------------------------------
extern "C" void kernel_launch(void* const* d_in, const int* in_sizes, int n_in,
                              void* d_out, int out_size, void* d_ws, size_t ws_size,
                              hipStream_t stream) {
    const float* af   = (const float*)d_in[0];
    const int*   idx  = (const int*)d_in[1];
    const float* W    = (const float*)d_in[2];
    const float* bias = (const float*)d_in[3];

    const int N = in_sizes[1];
    const int D = in_sizes[3] / 4;          // 128
    const int B = out_size / (2 * D);       // 16384
    float* E = (float*)d_out;               // d_out == carry_state_evolved

    char* ws = (char*)d_ws;
    int* seg = (int*)ws;
    size_t off = (((size_t)(B + 1) * sizeof(int)) + 255) & ~(size_t)255;
    float* carry = (float*)(ws + off);
    float* mem   = carry + (size_t)B * D;

    seg_starts_kernel<<<(B + 1 + 255) / 256, 256, 0, stream>>>(idx, N, B + 1, seg);

    const size_t zn = 2 * (size_t)B * D;    // carry + mem contiguous
    zero_kernel<<<(unsigned)((zn + 255) / 256), 256, 0, stream>>>(carry, zn);

    const int STEPS = 8;
    for (int s = 0; s < STEPS; ++s) {
        attn_kernel<<<B, 128, 0, stream>>>(af, seg, carry, E);
        if (s < STEPS - 1) {  // final LSTM update is dead code w.r.t. output
            dim3 grid(B / 32, D / 16);
            lstm_kernel<<<grid, 32, 0, stream>>>(E, W, bias, mem, carry, B, D);
        }
    }
}